// GCN_Surrogate_824633721177
// MI455X (gfx1250) — compile-verified
//
#include <hip/hip_runtime.h>
#include <hip/hip_bf16.h>

typedef float v2f __attribute__((ext_vector_type(2)));
typedef float v8f __attribute__((ext_vector_type(8)));

#define NNODES 50000
#define NEDGES 800000
#define HDIM   128

// ---------------------------------------------------------------------------
// Preprocessing kernels
// ---------------------------------------------------------------------------

__global__ __launch_bounds__(256) void gcn_set_deg1(float* deg, int n) {
    int i = blockIdx.x * blockDim.x + threadIdx.x;
    if (i < n) deg[i] = 1.0f;  // self-loop contributes 1 to every node
}

__global__ __launch_bounds__(256) void gcn_prep_edges(const long long* __restrict__ ei,
                                                      int* __restrict__ src32,
                                                      int* __restrict__ dst32,
                                                      float* __restrict__ deg, int e) {
    int i = blockIdx.x * blockDim.x + threadIdx.x;
    if (i >= e) return;
    int s = (int)ei[i];
    int d = (int)ei[(size_t)NEDGES + i];
    src32[i] = s;
    dst32[i] = d;
    atomicAdd(&deg[d], 1.0f);
}

__global__ __launch_bounds__(256) void gcn_dinv(const float* __restrict__ deg,
                                                float* __restrict__ dinv, int n) {
    int i = blockIdx.x * blockDim.x + threadIdx.x;
    if (i < n) dinv[i] = rsqrtf(deg[i]);  // deg >= 1 always (self loops)
}

__global__ __launch_bounds__(256) void gcn_norm(const int* __restrict__ src32,
                                                const int* __restrict__ dst32,
                                                const float* __restrict__ dinv,
                                                float* __restrict__ norm, int e) {
    int i = blockIdx.x * blockDim.x + threadIdx.x;
    if (i < e) norm[i] = dinv[src32[i]] * dinv[dst32[i]];
}

// ---------------------------------------------------------------------------
// WMMA fp32 GEMM:  out[M,128] = act(A[M,K]) @ W[128,K]^T (+ bias)
// One wave computes one 16x16 tile of out via V_WMMA_F32_16X16X4_F32.
// Block = 256 threads = 8 waves = one 16-row stripe (8 col tiles of 16).
//
// A-matrix 16x4 f32 layout (wave32): lane l holds row m = l%16;
//   lanes 0-15 hold K = {k, k+1}, lanes 16-31 hold K = {k+2, k+3}.
// B-matrix 4x16: lane l holds col n = l%16, same K split -> since B = W^T
//   with W row-major [128,K], the load pattern is identical to A's.
// C/D 16x16 f32: VGPR r -> row m0 + r + 8*(l>=16), col n0 + l%16.
//
// RELU is a compile-time parameter so the non-relu instantiation has a pure
// load->wmma inner loop and the relu one emits a single v_max_num per element
// (no per-element cndmask merging of both paths).
// ---------------------------------------------------------------------------

template <bool RELU>
__global__ __launch_bounds__(256) void gcn_gemm_wmma(const float* __restrict__ A,
                                                     const float* __restrict__ W,
                                                     const float* __restrict__ bias,
                                                     float* __restrict__ out,
                                                     int K) {
    const int lane  = threadIdx.x & 31;
    const int wave  = threadIdx.x >> 5;          // 0..7 -> column tile
    const int m0    = blockIdx.x * 16;
    const int n0    = wave * 16;
    const int mn    = lane & 15;
    const int khalf = (lane >> 4) * 2;           // 0 or 2

    const float* __restrict__ arow = A + (size_t)(m0 + mn) * K + khalf;
    const float* __restrict__ wrow = W + (size_t)(n0 + mn) * K + khalf;

    v8f c = {};
    for (int k = 0; k < K; k += 4) {
        v2f a = *(const v2f*)(arow + k);
        if (RELU) {
            a.x = fmaxf(a.x, 0.0f);
            a.y = fmaxf(a.y, 0.0f);
        }
        v2f b = *(const v2f*)(wrow + k);
        // 8 args: (neg_a, A, neg_b, B, c_mod, C, reuse_a, reuse_b)
        c = __builtin_amdgcn_wmma_f32_16x16x4_f32(false, a, false, b,
                                                  (short)0, c, false, false);
    }

    const int mbase = m0 + ((lane >> 4) << 3);   // +8 for upper lane half
    const int ncol  = n0 + mn;
    const float badd = bias ? bias[ncol] : 0.0f;
#pragma unroll
    for (int r = 0; r < 8; ++r) {
        out[(size_t)(mbase + r) * HDIM + ncol] = c[r] + badd;
    }
}

// ---------------------------------------------------------------------------
// agg[n,c] = bias[c] + dinv[n]^2 * hw[n,c]      (self-loop message + bias)
// ---------------------------------------------------------------------------
__global__ __launch_bounds__(256) void gcn_init_agg(const float* __restrict__ hw,
                                                    const float* __restrict__ dinv,
                                                    const float* __restrict__ bias,
                                                    float* __restrict__ agg, int total) {
    int i = blockIdx.x * blockDim.x + threadIdx.x;
    if (i >= total) return;
    int n = i >> 7;
    int cch = i & 127;
    float di = dinv[n];
    agg[i] = bias[cch] + di * di * hw[i];
}

// ---------------------------------------------------------------------------
// Scatter: one wave per edge; lane l handles channels 4l..4l+3 (float4 gather,
// 4 global_atomic_add_f32 into agg[dst]).
// ---------------------------------------------------------------------------
__global__ __launch_bounds__(256) void gcn_scatter(const float* __restrict__ hw,
                                                   const int* __restrict__ src,
                                                   const int* __restrict__ dst,
                                                   const float* __restrict__ norm,
                                                   float* __restrict__ agg, int e) {
    int edge = blockIdx.x * (blockDim.x >> 5) + (threadIdx.x >> 5);
    if (edge >= e) return;
    int lane = threadIdx.x & 31;
    int s = src[edge];
    int d = dst[edge];
    float nm = norm[edge];
    float4 v = ((const float4*)(hw + (size_t)s * HDIM))[lane];
    float* ag = agg + (size_t)d * HDIM + lane * 4;
    atomicAdd(ag + 0, v.x * nm);
    atomicAdd(ag + 1, v.y * nm);
    atomicAdd(ag + 2, v.z * nm);
    atomicAdd(ag + 3, v.w * nm);
}

// ---------------------------------------------------------------------------
// Decoder: out[n,0..2] = dec_b + relu(agg[n,:]) @ dec_w[3,128]^T
// One wave per node, float4 per lane, shuffle reduction.
// ---------------------------------------------------------------------------
__global__ __launch_bounds__(256) void gcn_decoder(const float* __restrict__ agg,
                                                   const float* __restrict__ dec_w,
                                                   const float* __restrict__ dec_b,
                                                   float* __restrict__ out, int n) {
    int node = blockIdx.x * (blockDim.x >> 5) + (threadIdx.x >> 5);
    if (node >= n) return;
    int lane = threadIdx.x & 31;
    float4 h = ((const float4*)(agg + (size_t)node * HDIM))[lane];
    h.x = fmaxf(h.x, 0.0f);
    h.y = fmaxf(h.y, 0.0f);
    h.z = fmaxf(h.z, 0.0f);
    h.w = fmaxf(h.w, 0.0f);
    float acc[3];
#pragma unroll
    for (int j = 0; j < 3; ++j) {
        float4 w = ((const float4*)(dec_w + j * HDIM))[lane];
        acc[j] = h.x * w.x + h.y * w.y + h.z * w.z + h.w * w.w;
    }
#pragma unroll
    for (int off = 16; off > 0; off >>= 1) {
#pragma unroll
        for (int j = 0; j < 3; ++j) acc[j] += __shfl_down(acc[j], off, 32);
    }
    if (lane == 0) {
#pragma unroll
        for (int j = 0; j < 3; ++j) out[(size_t)node * 3 + j] = acc[j] + dec_b[j];
    }
}

// ---------------------------------------------------------------------------
// Host-side launcher
// ---------------------------------------------------------------------------
extern "C" void kernel_launch(void* const* d_in, const int* in_sizes, int n_in,
                              void* d_out, int out_size, void* d_ws, size_t ws_size,
                              hipStream_t stream) {
    const float*     x      = (const float*)d_in[0];
    const long long* ei     = (const long long*)d_in[1];   // int64 edge_index [2,E]
    const float*     enc_w  = (const float*)d_in[2];       // [128,16]
    const float*     enc_b  = (const float*)d_in[3];       // [128]
    const float*     conv_w = (const float*)d_in[4];       // [3,128,128]
    const float*     conv_b = (const float*)d_in[5];       // [3,128]
    const float*     dec_w  = (const float*)d_in[6];       // [3,128]
    const float*     dec_b  = (const float*)d_in[7];       // [3]
    float* out = (float*)d_out;

    // Workspace carve-up (256B aligned)
    char* ws = (char*)d_ws;
    size_t off = 0;
    auto carve = [&](size_t bytes) {
        char* p = ws + off;
        off += (bytes + 255) & ~(size_t)255;
        return p;
    };
    float* deg   = (float*)carve((size_t)NNODES * 4);
    float* dinv  = (float*)carve((size_t)NNODES * 4);
    int*   src32 = (int*)  carve((size_t)NEDGES * 4);
    int*   dst32 = (int*)  carve((size_t)NEDGES * 4);
    float* normv = (float*)carve((size_t)NEDGES * 4);
    float* bufA  = (float*)carve((size_t)NNODES * HDIM * 4);
    float* bufHW = (float*)carve((size_t)NNODES * HDIM * 4);
    float* bufB  = (float*)carve((size_t)NNODES * HDIM * 4);

    const int TB = 256;
    const int nb_N  = (NNODES + TB - 1) / TB;
    const int nb_E  = (NEDGES + TB - 1) / TB;
    const int nb_NH = (NNODES * HDIM + TB - 1) / TB;
    const int nb_Ew = (NEDGES + 7) / 8;       // 8 waves (edges) per block
    const int nb_Nw = (NNODES + 7) / 8;       // 8 waves (nodes) per block
    const int nb_gemm = NNODES / 16;          // 50000 / 16 = 3125 row tiles

    // --- normalization precompute ---
    gcn_set_deg1<<<nb_N, TB, 0, stream>>>(deg, NNODES);
    gcn_prep_edges<<<nb_E, TB, 0, stream>>>(ei, src32, dst32, deg, NEDGES);
    gcn_dinv<<<nb_N, TB, 0, stream>>>(deg, dinv, NNODES);
    gcn_norm<<<nb_E, TB, 0, stream>>>(src32, dst32, dinv, normv, NEDGES);

    // --- encoder: h = x @ enc_w^T + enc_b  (K = 16, no input relu) ---
    gcn_gemm_wmma<false><<<nb_gemm, TB, 0, stream>>>(x, enc_w, enc_b, bufA, 16);

    // --- 3 GCN layers (relu folded into next GEMM's A-load) ---
    float* hcur = bufA;
    float* aggb = bufB;
    for (int layer = 0; layer < 3; ++layer) {
        const float* Wl = conv_w + (size_t)layer * HDIM * HDIM;
        const float* bl = conv_b + (size_t)layer * HDIM;
        if (layer == 0) {
            gcn_gemm_wmma<false><<<nb_gemm, TB, 0, stream>>>(hcur, Wl, nullptr,
                                                             bufHW, HDIM);
        } else {
            gcn_gemm_wmma<true><<<nb_gemm, TB, 0, stream>>>(hcur, Wl, nullptr,
                                                            bufHW, HDIM);
        }
        gcn_init_agg<<<nb_NH, TB, 0, stream>>>(bufHW, dinv, bl, aggb, NNODES * HDIM);
        gcn_scatter<<<nb_Ew, TB, 0, stream>>>(bufHW, src32, dst32, normv, aggb, NEDGES);
        // ping-pong: aggregate becomes next layer's input
        float* t = hcur; hcur = aggb; aggb = t;
    }

    // --- decoder (applies relu to final aggregate) ---
    gcn_decoder<<<nb_Nw, TB, 0, stream>>>(hcur, dec_w, dec_b, out, NNODES);
}